// DistanceLoss_58875411694215
// MI455X (gfx1250) — compile-verified
//
#include <hip/hip_runtime.h>
#include <hip/hip_bf16.h>

// DistanceLoss for MI455X / gfx1250 (wave32, WMMA).
// Pipeline:
//   K0: zero 4 accumulator floats in ws
//   K1: x-pass clipped squared DT (both targets: zeros and ones), LDS row cache
//   K2: y-pass (+dy^2, |dy|<=15)
//   K3: z-pass over batch (+dz^2, |dz|<=3) + combined = min(sqrt(f0)+sqrt(f1), 15)
//   K4: fp32 WMMA (v_wmma_f32_16x16x4_f32) diagonal-chunk dot products for
//       S1=sum(t*comb), inter=sum(p*t), Sp=sum(p), St=sum(t)
//   K5: finalize scalar loss
// Exactness: THRESH=15 clip makes +-15 (xy) / +-3 (z) windows exact (see analysis).

typedef __attribute__((ext_vector_type(2))) float v2f;
typedef __attribute__((ext_vector_type(8))) float v8f;

#define BATCH  4
#define YDIM   512
#define XDIM   512
#define NTOT   (BATCH * YDIM * XDIM)   // 1048576 = 16384 * 64
#define RWIN   15
#define BIGSQ  1.0e9f
#define THRESH 15.0f

__global__ void zero_sums_kernel(float* __restrict__ sums) {
    if (threadIdx.x < 4) sums[threadIdx.x] = 0.0f;
}

// K1: per-row (b*Y+y) 1D clipped squared DT along x, for both target sets.
__global__ void edt_xpass(const float* __restrict__ pred,
                          float* __restrict__ a0, float* __restrict__ a1) {
    __shared__ float row[XDIM];
    const int r    = blockIdx.x;        // r = b*YDIM + y
    const int x    = threadIdx.x;
    const int base = r * XDIM;
    row[x] = pred[base + x];
    __syncthreads();
    float d0 = BIGSQ, d1 = BIGSQ;       // d0: to nearest 0, d1: to nearest 1
    #pragma unroll
    for (int dx = -RWIN; dx <= RWIN; ++dx) {
        int xx = x + dx;
        if (xx >= 0 && xx < XDIM) {
            float v  = row[xx];
            float dd = (float)(dx * dx);
            if (v == 0.0f) d0 = fminf(d0, dd);
            else           d1 = fminf(d1, dd);
        }
    }
    a0[base + x] = d0;
    a1[base + x] = d1;
}

// K2: min over dy of a[y+dy] + dy^2 (within the same batch slice).
__global__ void edt_ypass(const float* __restrict__ a0, const float* __restrict__ a1,
                          float* __restrict__ b0, float* __restrict__ b1) {
    const int x = blockIdx.x * blockDim.x + threadIdx.x;
    const int y = blockIdx.y;
    const int b = blockIdx.z;
    float d0 = BIGSQ, d1 = BIGSQ;
    #pragma unroll
    for (int dy = -RWIN; dy <= RWIN; ++dy) {
        int yy = y + dy;
        if (yy >= 0 && yy < YDIM) {
            int   idx = (b * YDIM + yy) * XDIM + x;
            float dd  = (float)(dy * dy);
            d0 = fminf(d0, a0[idx] + dd);
            d1 = fminf(d1, a1[idx] + dd);
        }
    }
    const int o = (b * YDIM + y) * XDIM + x;
    b0[o] = d0;
    b1[o] = d1;
}

// K3: min over batch axis (+dz^2) and fuse the clip/sqrt combine.
__global__ void edt_zpass_combine(const float* __restrict__ b0, const float* __restrict__ b1,
                                  float* __restrict__ comb) {
    const int x = blockIdx.x * blockDim.x + threadIdx.x;
    const int y = blockIdx.y;
    const int b = blockIdx.z;
    float f0 = BIGSQ, f1 = BIGSQ;
    #pragma unroll
    for (int bb = 0; bb < BATCH; ++bb) {
        float dd  = (float)((bb - b) * (bb - b));
        int   idx = (bb * YDIM + y) * XDIM + x;
        f0 = fminf(f0, b0[idx] + dd);
        f1 = fminf(f1, b1[idx] + dd);
    }
    comb[(b * YDIM + y) * XDIM + x] = fminf(sqrtf(f0) + sqrtf(f1), THRESH);
}

// K4: fp32 WMMA reduction. Each 16x16x4 WMMA consumes a 64-element chunk:
//   A[m][k] = x[base + 4m + k], B[k][m] = y[base + 4m + k] -> D[m][m] = chunk dot.
// ISA layouts make both operands the same per-lane float2 gather at
// off = 4*(lane&15) + 2*(lane>>4).
__global__ void wmma_reduce(const float* __restrict__ comb,
                            const float* __restrict__ pred,
                            const float* __restrict__ truth,
                            float* __restrict__ sums) {
    const int tid     = blockIdx.x * blockDim.x + threadIdx.x;
    const int lane    = threadIdx.x & 31;
    const int wave    = tid >> 5;
    const int nwaves  = (gridDim.x * blockDim.x) >> 5;
    const int nchunks = NTOT / 64;                 // divides exactly
    const int loff    = 4 * (lane & 15) + 2 * (lane >> 4);

    v8f acc_s1 = {}; v8f acc_in = {}; v8f acc_sp = {}; v8f acc_st = {};
    const v2f one2 = {1.0f, 1.0f};

    for (int c = wave; c < nchunks; c += nwaves) {   // wave-uniform: EXEC stays all-ones
        const int off = c * 64 + loff;
        v2f c2 = *(const v2f*)(comb  + off);
        v2f p2 = *(const v2f*)(pred  + off);
        v2f t2 = *(const v2f*)(truth + off);
        acc_s1 = __builtin_amdgcn_wmma_f32_16x16x4_f32(false, c2, false, t2,   (short)0, acc_s1, false, false);
        acc_in = __builtin_amdgcn_wmma_f32_16x16x4_f32(false, p2, false, t2,   (short)0, acc_in, false, false);
        acc_sp = __builtin_amdgcn_wmma_f32_16x16x4_f32(false, p2, false, one2, (short)0, acc_sp, false, false);
        acc_st = __builtin_amdgcn_wmma_f32_16x16x4_f32(false, t2, false, one2, (short)0, acc_st, false, false);
    }

    // Diagonal D[m][m]: m<8 -> VGPR m, lane m ; m>=8 -> VGPR m-8, lane m+16 (lanes 24..31).
    float v0 = 0.0f, v1 = 0.0f, v2 = 0.0f, v3 = 0.0f;
    #pragma unroll
    for (int j = 0; j < 8; ++j) {
        const bool sel = (lane == j) || (lane == j + 24);
        v0 += sel ? acc_s1[j] : 0.0f;
        v1 += sel ? acc_in[j] : 0.0f;
        v2 += sel ? acc_sp[j] : 0.0f;
        v3 += sel ? acc_st[j] : 0.0f;
    }
    #pragma unroll
    for (int o = 16; o > 0; o >>= 1) {
        v0 += __shfl_down(v0, o, 32);
        v1 += __shfl_down(v1, o, 32);
        v2 += __shfl_down(v2, o, 32);
        v3 += __shfl_down(v3, o, 32);
    }
    if (lane == 0) {
        atomicAdd(&sums[0], v0);
        atomicAdd(&sums[1], v1);
        atomicAdd(&sums[2], v2);
        atomicAdd(&sums[3], v3);
    }
}

__global__ void finalize_kernel(const float* __restrict__ sums, float* __restrict__ out) {
    const float s1    = sums[0];
    const float inter = sums[1];
    const float sp    = sums[2];
    const float st    = sums[3];             // y_true is binary -> mask count == St
    const float n         = fmaxf(st, 1.0f);
    const float dist_mean = s1 / n;
    const float dice      = (2.0f * inter + 1e-6f) / (sp + st + 1e-6f);
    out[0] = (1.0f - dist_mean) + (1.0f - dice);
}

extern "C" void kernel_launch(void* const* d_in, const int* in_sizes, int n_in,
                              void* d_out, int out_size, void* d_ws, size_t ws_size,
                              hipStream_t stream) {
    (void)in_sizes; (void)n_in; (void)out_size; (void)ws_size;
    const float* pred  = (const float*)d_in[0];   // y_pred, (4,1,512,512) f32
    const float* truth = (const float*)d_in[1];   // y_true, (4,1,512,512) f32
    float*       out   = (float*)d_out;

    // Workspace: 4 DT buffers (16 MB) + 4 accumulator floats; comb reuses a0.
    float* ws   = (float*)d_ws;
    float* a0   = ws;                 // x-pass DT^2 to zeros
    float* a1   = ws + 1 * NTOT;      // x-pass DT^2 to ones
    float* b0   = ws + 2 * NTOT;      // y-pass DT^2 to zeros
    float* b1   = ws + 3 * NTOT;      // y-pass DT^2 to ones
    float* comb = a0;                 // a0 dead after K2 -> reuse for combined
    float* sums = ws + 4 * NTOT;      // 4 f32 accumulators

    zero_sums_kernel<<<1, 32, 0, stream>>>(sums);
    edt_xpass<<<BATCH * YDIM, XDIM, 0, stream>>>(pred, a0, a1);
    edt_ypass<<<dim3(XDIM / 256, YDIM, BATCH), 256, 0, stream>>>(a0, a1, b0, b1);
    edt_zpass_combine<<<dim3(XDIM / 256, YDIM, BATCH), 256, 0, stream>>>(b0, b1, comb);
    wmma_reduce<<<512, 256, 0, stream>>>(comb, pred, truth, sums);
    finalize_kernel<<<1, 32, 0, stream>>>(sums, out);
}